// UniHeadSimple_66692252172800
// MI455X (gfx1250) — compile-verified
//
#include <hip/hip_runtime.h>
#include <hip/hip_bf16.h>

typedef __attribute__((ext_vector_type(2))) float v2f;
typedef __attribute__((ext_vector_type(8))) float v8f;

static constexpr int BATCH = 64;
static constexpr int IMG_W = 512;
static constexpr int IMG_H = 512;
static constexpr int NPIX  = IMG_W * IMG_H;      // 262144
static constexpr int NB    = 64;                 // blocks per sample
static constexpr int TPB   = 256;
static constexpr int PIX_PER_BLOCK = NPIX / NB;  // 4096
static constexpr int VEC_ITERS = PIX_PER_BLOCK / (TPB * 4); // 4 float4 per thread

__device__ __forceinline__ int imin(int a, int b) { return a < b ? a : b; }
__device__ __forceinline__ int imax(int a, int b) { return a > b ? a : b; }

// ---------------- Stage 1: streaming fused reduction over the masks -------
__global__ __launch_bounds__(TPB) void seg_loss_stage1(
    const float* __restrict__ pm,   // [B,1,H,W] logits
    const int*   __restrict__ tm,   // [B,H,W] targets {0,1}
    const float* __restrict__ pb,   // [B,4] pred bbox (xywh, normalized)
    float* __restrict__ wsf,        // [4][BATCH][NB] float partials
    int*   __restrict__ wsi)        // [6][BATCH][NB] int partials
{
    const int s   = blockIdx.y;
    const int b   = blockIdx.x;
    const int tid = threadIdx.x;

    // Predicted box -> clipped integer pixel bounds (uniform per sample).
    const float cx = pb[s*4+0], cy = pb[s*4+1], bw = pb[s*4+2], bh = pb[s*4+3];
    const float bx1 = cx - bw*0.5f, by1 = cy - bh*0.5f;
    const float bx2 = cx + bw*0.5f, by2 = cy + bh*0.5f;
    const int x1p = imin(imax((int)(bx1 * (float)IMG_W), 0), IMG_W);
    const int x2p = imin(imax((int)(bx2 * (float)IMG_W), 0), IMG_W);
    const int y1p = imin(imax((int)(by1 * (float)IMG_H), 0), IMG_H);
    const int y2p = imin(imax((int)(by2 * (float)IMG_H), 0), IMG_H);

    const size_t base = (size_t)s * NPIX + (size_t)b * PIX_PER_BLOCK;
    const float4* pm4 = (const float4*)(pm + base);
    const int4*   tm4 = (const int4*)(tm + base);

    float s_pt = 0.f, s_p = 0.f, s_t = 0.f, s_bce = 0.f;
    int area = 0, inter = 0;
    int mnx = IMG_W, mny = IMG_H, mxx = -1, mxy = -1;

    #pragma unroll
    for (int i = 0; i < VEC_ITERS; ++i) {
        const int v = i * TPB + tid;          // float4 index inside chunk
        const float4 xv = pm4[v];
        const int4   tv = tm4[v];
        const int pix  = b * PIX_PER_BLOCK + v * 4;
        const int row  = pix >> 9;            // / IMG_W
        const int col0 = pix & (IMG_W - 1);
        const bool iny = (row >= y1p) && (row < y2p);
        const float xs[4] = {xv.x, xv.y, xv.z, xv.w};
        const int   ts[4] = {tv.x, tv.y, tv.z, tv.w};
        #pragma unroll
        for (int j = 0; j < 4; ++j) {
            const float x = xs[j];
            const float t = (float)ts[j];
            const float ex = expf(-fabsf(x));            // shared exp
            const float p  = (x >= 0.f) ? 1.f/(1.f+ex) : ex/(1.f+ex);
            s_p  += p;
            s_t  += t;
            s_pt += p * t;
            s_bce += fmaxf(x, 0.f) - x * t + log1pf(ex); // stable BCE
            if (x > 0.f) {                               // probs > 0.5
                const int col = col0 + j;
                ++area;
                mnx = imin(mnx, col); mxx = imax(mxx, col);
                mny = imin(mny, row); mxy = imax(mxy, row);
                if (iny && col >= x1p && col < x2p) ++inter;
            }
        }
    }

    // wave32 shuffle tree (deterministic)
    const int lane = tid & 31, wave = tid >> 5;
    #pragma unroll
    for (int off = 16; off > 0; off >>= 1) {
        s_pt  += __shfl_xor(s_pt,  off, 32);
        s_p   += __shfl_xor(s_p,   off, 32);
        s_t   += __shfl_xor(s_t,   off, 32);
        s_bce += __shfl_xor(s_bce, off, 32);
        area  += __shfl_xor(area,  off, 32);
        inter += __shfl_xor(inter, off, 32);
        mnx = imin(mnx, __shfl_xor(mnx, off, 32));
        mny = imin(mny, __shfl_xor(mny, off, 32));
        mxx = imax(mxx, __shfl_xor(mxx, off, 32));
        mxy = imax(mxy, __shfl_xor(mxy, off, 32));
    }

    __shared__ float sf[4][8];
    __shared__ int   si[6][8];
    if (lane == 0) {
        sf[0][wave] = s_pt; sf[1][wave] = s_p; sf[2][wave] = s_t; sf[3][wave] = s_bce;
        si[0][wave] = area; si[1][wave] = inter;
        si[2][wave] = mnx;  si[3][wave] = mny; si[4][wave] = mxx; si[5][wave] = mxy;
    }
    __syncthreads();
    if (tid == 0) {
        float f0 = 0.f, f1 = 0.f, f2 = 0.f, f3 = 0.f;
        int a = 0, it = 0, x1 = IMG_W, y1 = IMG_H, x2 = -1, y2 = -1;
        #pragma unroll
        for (int w2 = 0; w2 < TPB/32; ++w2) {
            f0 += sf[0][w2]; f1 += sf[1][w2]; f2 += sf[2][w2]; f3 += sf[3][w2];
            a  += si[0][w2]; it += si[1][w2];
            x1 = imin(x1, si[2][w2]); y1 = imin(y1, si[3][w2]);
            x2 = imax(x2, si[4][w2]); y2 = imax(y2, si[5][w2]);
        }
        wsf[(0*BATCH + s)*NB + b] = f0;
        wsf[(1*BATCH + s)*NB + b] = f1;
        wsf[(2*BATCH + s)*NB + b] = f2;
        wsf[(3*BATCH + s)*NB + b] = f3;
        wsi[(0*BATCH + s)*NB + b] = a;
        wsi[(1*BATCH + s)*NB + b] = it;
        wsi[(2*BATCH + s)*NB + b] = x1;
        wsi[(3*BATCH + s)*NB + b] = y1;
        wsi[(4*BATCH + s)*NB + b] = x2;
        wsi[(5*BATCH + s)*NB + b] = y2;
    }
}

// ---------------- Stage 2: WMMA cross-block reduction + finalize -----------
__device__ __forceinline__ float giou_xyxy(const float* a, const float* b) {
    const float ix1 = fmaxf(a[0], b[0]), iy1 = fmaxf(a[1], b[1]);
    const float ix2 = fminf(a[2], b[2]), iy2 = fminf(a[3], b[3]);
    const float inter = fmaxf(ix2 - ix1, 0.f) * fmaxf(iy2 - iy1, 0.f);
    const float area_a = fmaxf(a[2] - a[0], 0.f) * fmaxf(a[3] - a[1], 0.f);
    const float area_b = fmaxf(b[2] - b[0], 0.f) * fmaxf(b[3] - b[1], 0.f);
    const float uni = area_a + area_b - inter;
    const float iou = inter / fmaxf(uni, 1e-6f);
    const float cx1 = fminf(a[0], b[0]), cy1 = fminf(a[1], b[1]);
    const float cx2 = fmaxf(a[2], b[2]), cy2 = fmaxf(a[3], b[3]);
    const float c = fmaxf(cx2 - cx1, 0.f) * fmaxf(cy2 - cy1, 0.f);
    return iou - (c - uni) / fmaxf(c, 1e-6f);
}

__device__ __forceinline__ float box_loss_xywh(const float* p, const float* t) {
    const float l1 = fabsf(p[0]-t[0]) + fabsf(p[1]-t[1]) + fabsf(p[2]-t[2]) + fabsf(p[3]-t[3]);
    const float pa[4] = {p[0]-p[2]*0.5f, p[1]-p[3]*0.5f, p[0]+p[2]*0.5f, p[1]+p[3]*0.5f};
    const float tb[4] = {t[0]-t[2]*0.5f, t[1]-t[3]*0.5f, t[0]+t[2]*0.5f, t[1]+t[3]*0.5f};
    return l1 + (1.0f - giou_xyxy(pa, tb));
}

__global__ __launch_bounds__(TPB) void seg_loss_stage2(
    const float* __restrict__ wsf,  // [4][BATCH][NB]
    const int*   __restrict__ wsi,  // [6][BATCH][NB]
    const float* __restrict__ pb,   // [B,4]
    const float* __restrict__ gb,   // [B,4]
    float* __restrict__ out)
{
    __shared__ float ssum[4][BATCH];
    __shared__ float contrib[BATCH];

    const int tid  = threadIdx.x;
    const int wave = tid >> 5;
    const int lane = tid & 31;

    // --- Exact f32 cross-block sums via V_WMMA_F32_16X16X4_F32, B = ones ---
    // 8 waves: quantity q = wave&3, row-groups {(wave>>2)*2, +1}. Each WMMA
    // sums 4 block-partials for 16 samples; C chains the f32 accumulation.
    {
        const int q = wave & 3;
        const int rg0 = (wave >> 2) * 2;
        const int m  = lane & 15;      // sample row within group
        const int hf = lane >> 4;      // lane-half selects K pair {0,1} vs {2,3}
        for (int rg = rg0; rg < rg0 + 2; ++rg) {
            const int srow = rg * 16 + m;
            const float* src = wsf + ((size_t)q * BATCH + srow) * NB;
            v8f c = {};
            v2f bones; bones.x = 1.0f; bones.y = 1.0f;
            #pragma unroll 4
            for (int ch = 0; ch < NB / 4; ++ch) {
                v2f a;
                a.x = src[ch*4 + hf*2 + 0];   // A[m][hf*2]
                a.y = src[ch*4 + hf*2 + 1];   // A[m][hf*2+1]
                c = __builtin_amdgcn_wmma_f32_16x16x4_f32(
                        false, a, false, bones, (short)0, c, false, false);
            }
            // D[m][n] == row sum for all n. Lane 0 holds M=0..7 in VGPRs 0..7,
            // lane 16 holds M=8..15.
            if (lane == 0) {
                #pragma unroll
                for (int j = 0; j < 8; ++j) ssum[q][rg*16 + j] = c[j];
            } else if (lane == 16) {
                #pragma unroll
                for (int j = 0; j < 8; ++j) ssum[q][rg*16 + 8 + j] = c[j];
            }
        }
    }
    __syncthreads();

    if (tid < BATCH) {
        const int s = tid;
        // integer partial reduction (fixed order -> deterministic)
        int area = 0, inter = 0, mnx = IMG_W, mny = IMG_H, mxx = -1, mxy = -1;
        for (int b = 0; b < NB; ++b) {
            area += wsi[(0*BATCH + s)*NB + b];
            inter+= wsi[(1*BATCH + s)*NB + b];
            mnx = imin(mnx, wsi[(2*BATCH + s)*NB + b]);
            mny = imin(mny, wsi[(3*BATCH + s)*NB + b]);
            mxx = imax(mxx, wsi[(4*BATCH + s)*NB + b]);
            mxy = imax(mxy, wsi[(5*BATCH + s)*NB + b]);
        }
        const float s_pt = ssum[0][s], s_p = ssum[1][s];
        const float s_t  = ssum[2][s], s_bce = ssum[3][s];

        // dice
        const float dice = 1.f - (2.f * s_pt + 1.f) / (s_p + s_t + 1.f);
        // seg->box IoU
        const float s2b = (area > 0) ? ((float)inter / fmaxf((float)area, 1.f)) : 0.f;

        // detection loss vs synthetic GT box
        const float g0 = gb[s*4+0], g1 = gb[s*4+1], g2 = gb[s*4+2], g3 = gb[s*4+3];
        const float gx1 = g0 * (0.45f * (float)IMG_W);
        const float gy1 = g1 * (0.45f * (float)IMG_H);
        const float gx2 = gx1 + 1.f + g2 * (0.5f * (float)IMG_W);
        const float gy2 = gy1 + 1.f + g3 * (0.5f * (float)IMG_H);
        const float gtn[4] = { (gx1+gx2)*0.5f/(float)IMG_H, (gy1+gy2)*0.5f/(float)IMG_W,
                               (gx2-gx1)/(float)IMG_H,      (gy2-gy1)/(float)IMG_W };
        const float pr[4] = { pb[s*4+0], pb[s*4+1], pb[s*4+2], pb[s*4+3] };
        const float det = box_loss_xywh(pr, gtn);

        // box->seg loss: pred box xywh->xyxy->xywh roundtrip (as in reference)
        const float bx1 = pr[0]-pr[2]*0.5f, by1 = pr[1]-pr[3]*0.5f;
        const float bx2 = pr[0]+pr[2]*0.5f, by2 = pr[1]+pr[3]*0.5f;
        const float pback[4] = { (bx1+bx2)*0.5f, (by1+by2)*0.5f, bx2-bx1, by2-by1 };
        float ox[4];
        if (area > 0) {
            ox[0] = (float)mnx / (float)IMG_W; ox[1] = (float)mny / (float)IMG_H;
            ox[2] = (float)mxx / (float)IMG_W; ox[3] = (float)mxy / (float)IMG_H;
        } else {
            ox[0] = ox[1] = ox[2] = ox[3] = 0.f;
        }
        const float oxywh[4] = { (ox[0]+ox[2])*0.5f, (ox[1]+ox[3])*0.5f,
                                 ox[2]-ox[0],        ox[3]-ox[1] };
        const float b2s = box_loss_xywh(oxywh, pback);

        contrib[s] = (dice + (1.f - s2b) + b2s + 0.025f * det) * (1.f / (float)BATCH)
                   + s_bce * (1.f / ((float)BATCH * (float)NPIX));
    }
    __syncthreads();
    if (tid == 0) {
        float tot = 0.f;
        for (int i = 0; i < BATCH; ++i) tot += contrib[i];
        out[0] = tot;
    }
}

extern "C" void kernel_launch(void* const* d_in, const int* in_sizes, int n_in,
                              void* d_out, int out_size, void* d_ws, size_t ws_size,
                              hipStream_t stream) {
    const float* pred_mask  = (const float*)d_in[0];
    const float* pred_bbox  = (const float*)d_in[1];
    const float* gt_bbox    = (const float*)d_in[2];
    const int*   target     = (const int*)d_in[3];
    (void)n_in; (void)out_size; (void)ws_size;

    const int B = in_sizes[1] / 4;   // 64

    float* wsf = (float*)d_ws;                        // [4][BATCH][NB]
    int*   wsi = (int*)((float*)d_ws + 4 * BATCH * NB); // [6][BATCH][NB]

    seg_loss_stage1<<<dim3(NB, B), TPB, 0, stream>>>(pred_mask, target, pred_bbox, wsf, wsi);
    seg_loss_stage2<<<1, TPB, 0, stream>>>(wsf, wsi, pred_bbox, gt_bbox, (float*)d_out);
}